// LightGCN_17265768530449
// MI455X (gfx1250) — compile-verified
//
#include <hip/hip_runtime.h>
#include <stdint.h>

// ---------------------------------------------------------------------------
// LightGCN on MI455X (gfx1250, wave32)
//   out = (X + AX + A^2 X + A^3 X) / 4,  A: COO (4.8M nnz), X: 150000 x 64 f32
//
// Roofline: ~0.61 GFLOP/layer vs 57.6 MB/layer HBM edge stream; dense operand
// (38.4 MB) is L2-resident (192 MB L2), so gather/scatter stays on-die and the
// limiter is L2 f32-atomic throughput. No dense matmul exists -> no WMMA; the
// CDNA5-specific paths used are async global->LDS edge staging (ASYNCcnt) and
// native global_atomic_add_f32 at device scope.
// ---------------------------------------------------------------------------

#define BLOCK 256
#define TILE_EDGES 1024
#define WAVES_PER_BLOCK 8
#define EDGES_PER_WAVE (TILE_EDGES / WAVES_PER_BLOCK) // 128

// Low 32 bits of a generic (flat) LDS address are the LDS byte offset
// (aperture-based addrspacecast: high half = SHARED_BASE, low half = offset).
__device__ __forceinline__ uint32_t lds_addr_of(const void* p) {
    return (uint32_t)(uintptr_t)p;
}

// CDNA5 async copy: copies global memory at vaddr into LDS at offset vdst, tracked with ASYNCcnt.
__device__ __forceinline__ void async_load_lds_b128(uint32_t lds, const void* g) {
    asm volatile("global_load_async_to_lds_b128 %0, %1, off"
                 :: "v"(lds), "v"(g) : "memory");
}
__device__ __forceinline__ void async_load_lds_b32(uint32_t lds, const void* g) {
    asm volatile("global_load_async_to_lds_b32 %0, %1, off"
                 :: "v"(lds), "v"(g) : "memory");
}
__device__ __forceinline__ void wait_asynccnt0() {
    asm volatile("s_wait_asynccnt 0x0" ::: "memory");
}

// Native f32 atomic add (no return -> STOREcnt), device scope so the RMW is
// performed by the L2 atomic units and visible to subsequent kernels.
__device__ __forceinline__ void gatomic_add_f32(float* p, float v) {
    asm volatile("global_atomic_add_f32 %0, %1, off scope:SCOPE_DEV"
                 :: "v"(p), "v"(v) : "memory");
}

// ---------------------------------------------------------------------------
// init: cur = concat(user,item); out = 0.25*cur; next = 0   (float4 vectorized)
// ---------------------------------------------------------------------------
__global__ void lgcn_init(const float4* __restrict__ ue, const float4* __restrict__ ie,
                          float4* __restrict__ cur, float4* __restrict__ nxt,
                          float4* __restrict__ out, int nuser4, int ntot4) {
    int i = blockIdx.x * blockDim.x + threadIdx.x;
    if (i >= ntot4) return;
    float4 v = (i < nuser4) ? ue[i] : ie[i - nuser4];
    cur[i] = v;
    nxt[i] = make_float4(0.f, 0.f, 0.f, 0.f);
    out[i] = make_float4(0.25f * v.x, 0.25f * v.y, 0.25f * v.z, 0.25f * v.w);
}

// ---------------------------------------------------------------------------
// SpMM: y[row[e], :] += val[e] * x[col[e], :]
//   - block stages TILE_EDGES edges of (val,row,col) into LDS via async b128
//   - each of the 8 waves owns 128 staged edges; per edge: uniform LDS
//     broadcast of metadata, float2 gather from x (L2-resident), two f32
//     atomics into y (L2 atomic units)
// ---------------------------------------------------------------------------
__global__ void lgcn_spmm(const float* __restrict__ vals, const int* __restrict__ rows,
                          const int* __restrict__ cols, const float* __restrict__ x,
                          float* __restrict__ y, int nnz) {
    __shared__ float s_val[TILE_EDGES];
    __shared__ int   s_row[TILE_EDGES];
    __shared__ int   s_col[TILE_EDGES];

    const int t    = threadIdx.x;
    const int lane = t & 31;
    const int wv   = t >> 5;

    for (int base = blockIdx.x * TILE_EDGES; base < nnz;
         base += gridDim.x * TILE_EDGES) {
        // ---- stage: each thread copies 4 consecutive edges (16B) per array
        const int e4 = t * 4;
        const int ge = base + e4;
        if (ge + 4 <= nnz) {
            async_load_lds_b128(lds_addr_of(&s_val[e4]), vals + ge);
            async_load_lds_b128(lds_addr_of(&s_row[e4]), rows + ge);
            async_load_lds_b128(lds_addr_of(&s_col[e4]), cols + ge);
        } else if (ge < nnz) {   // ragged tail (nnz % 4 != 0)
            for (int k = 0; k < nnz - ge; ++k) {
                async_load_lds_b32(lds_addr_of(&s_val[e4 + k]), vals + ge + k);
                async_load_lds_b32(lds_addr_of(&s_row[e4 + k]), rows + ge + k);
                async_load_lds_b32(lds_addr_of(&s_col[e4 + k]), cols + ge + k);
            }
        }
        wait_asynccnt0();   // this wave's async copies landed in LDS
        __syncthreads();    // everyone's copies visible block-wide

        // ---- process: wave wv handles staged edges [wv*128, wv*128+128)
        int lim = nnz - base;
        if (lim > TILE_EDGES) lim = TILE_EDGES;
        const int start = wv * EDGES_PER_WAVE;
        const int end   = (start + EDGES_PER_WAVE < lim) ? start + EDGES_PER_WAVE : lim;

        for (int i = start; i < end; ++i) {
            const float v = s_val[i];          // uniform LDS broadcast
            const int   r = s_row[i];
            const int   c = s_col[i];
            const float2 xv = *(const float2*)(x + ((size_t)c << 6) + (lane << 1));
            float* dst = y + ((size_t)r << 6) + (lane << 1);
            gatomic_add_f32(dst,     v * xv.x);
            gatomic_add_f32(dst + 1, v * xv.y);
        }
        __syncthreads();    // LDS reuse fence before next tile restage
    }
}

// ---------------------------------------------------------------------------
// accumulate: out += 0.25*next ; optionally zero the other ping-pong buffer
// (which becomes the scatter target of the following layer)
// ---------------------------------------------------------------------------
__global__ void lgcn_accum(float4* __restrict__ out, const float4* __restrict__ nxt,
                           float4* __restrict__ zbuf, int do_zero, int ntot4) {
    int i = blockIdx.x * blockDim.x + threadIdx.x;
    if (i >= ntot4) return;
    float4 o = out[i];
    float4 a = nxt[i];
    o.x += 0.25f * a.x; o.y += 0.25f * a.y;
    o.z += 0.25f * a.z; o.w += 0.25f * a.w;
    out[i] = o;
    if (do_zero) zbuf[i] = make_float4(0.f, 0.f, 0.f, 0.f);
}

// ---------------------------------------------------------------------------
extern "C" void kernel_launch(void* const* d_in, const int* in_sizes, int n_in,
                              void* d_out, int out_size, void* d_ws, size_t ws_size,
                              hipStream_t stream) {
    (void)n_in; (void)out_size; (void)ws_size;

    const float* user_emb = (const float*)d_in[0];   // (100000, 64) f32
    const float* item_emb = (const float*)d_in[1];   // ( 50000, 64) f32
    const int*   adj_row  = (const int*)  d_in[2];   // (NNZ,) i32
    const int*   adj_col  = (const int*)  d_in[3];   // (NNZ,) i32
    const float* adj_val  = (const float*)d_in[4];   // (NNZ,) f32
    float*       out      = (float*)d_out;           // (150000, 64) f32 (users||items)

    const int DIM    = 64;
    const int nuser  = in_sizes[0] / DIM;            // 100000
    const int ntot   = nuser + in_sizes[1] / DIM;    // 150000
    const int nnz    = in_sizes[2];                  // 4800000
    const int ntot4  = ntot * DIM / 4;               // 2.4M float4
    const int nuser4 = nuser * DIM / 4;

    float* cur = (float*)d_ws;                       // 38.4 MB
    float* nxt = cur + (size_t)ntot * DIM;           // 38.4 MB (76.8 MB total)

    const int gridDense = (ntot4 + BLOCK - 1) / BLOCK;
    const int gridSpmm  = (nnz + TILE_EDGES - 1) / TILE_EDGES;   // 4688 blocks

    lgcn_init<<<gridDense, BLOCK, 0, stream>>>(
        (const float4*)user_emb, (const float4*)item_emb,
        (float4*)cur, (float4*)nxt, (float4*)out, nuser4, ntot4);

    for (int layer = 0; layer < 3; ++layer) {
        lgcn_spmm<<<gridSpmm, BLOCK, 0, stream>>>(adj_val, adj_row, adj_col,
                                                  cur, nxt, nnz);
        // out += 0.25*nxt; zero old cur (becomes next layer's scatter target)
        lgcn_accum<<<gridDense, BLOCK, 0, stream>>>(
            (float4*)out, (const float4*)nxt, (float4*)cur,
            (layer < 2) ? 1 : 0, ntot4);
        float* tmp = cur; cur = nxt; nxt = tmp;
    }
}